// GCN_7928509628812
// MI455X (gfx1250) — compile-verified
//
#include <hip/hip_runtime.h>

typedef __attribute__((ext_vector_type(2))) float v2f;
typedef __attribute__((ext_vector_type(8))) float v8f;

#define DIM 64  // D_IN == D_OUT == 64

// ---------------------------------------------------------------------------
// Zero-init the output (harness poisons d_out with 0xAA).
// ---------------------------------------------------------------------------
__global__ void gcn_zero_kernel(float* __restrict__ out, int n4) {
    int i = blockIdx.x * blockDim.x + threadIdx.x;
    if (i < n4) {
        float4 z = make_float4(0.f, 0.f, 0.f, 0.f);
        ((float4*)out)[i] = z;
    }
}

// ---------------------------------------------------------------------------
// support = X @ W  via V_WMMA_F32_16X16X4_F32 (fp32 matrix pipe, exact fp32).
// One wave per 16-row tile of X; 4 accumulators tile the 64 output columns.
// K-loop: 64 in steps of 4  -> 16 * 4 = 64 WMMA ops per wave.
//
// A 16x4 layout : lane l (l&15 = M row, l>>4 = K half), v[0]=A[m][2h], v[1]=A[m][2h+1]
// B 4x16 layout : lane l (l&15 = N col, l>>4 = K half), v[0]=B[2h][n], v[1]=B[2h+1][n]
// C/D 16x16     : VGPR v, lanes 0-15 -> M=v, lanes 16-31 -> M=v+8, N=l&15
// ---------------------------------------------------------------------------
__global__ void gcn_gemm_wmma_kernel(const float* __restrict__ X,
                                     const float* __restrict__ W,
                                     float* __restrict__ S,
                                     int n_row_tiles) {
    const int wave = threadIdx.x >> 5;
    const int lane = threadIdx.x & 31;
    const int tile = blockIdx.x * (blockDim.x >> 5) + wave;
    if (tile >= n_row_tiles) return;   // wave-uniform: EXEC stays all-ones

    const int m0    = tile << 4;       // first row of this 16-row tile
    const int mrow  = lane & 15;
    const int khalf = lane >> 4;       // 0 or 1
    const int ncol  = lane & 15;

    v8f acc0 = {}, acc1 = {}, acc2 = {}, acc3 = {};

    const float* __restrict__ xrow = X + (long long)(m0 + mrow) * DIM;

    #pragma unroll
    for (int k0 = 0; k0 < DIM; k0 += 4) {
        const int ka = k0 + (khalf << 1);          // this lane's K pair base
        v2f a = *(const v2f*)(xrow + ka);          // A[m][ka], A[m][ka+1]

        const float* __restrict__ w0 = W + ka * DIM;        // row ka   of W
        const float* __restrict__ w1 = W + (ka + 1) * DIM;  // row ka+1 of W

        v2f b0, b1, b2, b3;
        b0.x = w0[ 0 + ncol]; b0.y = w1[ 0 + ncol];
        b1.x = w0[16 + ncol]; b1.y = w1[16 + ncol];
        b2.x = w0[32 + ncol]; b2.y = w1[32 + ncol];
        b3.x = w0[48 + ncol]; b3.y = w1[48 + ncol];

        acc0 = __builtin_amdgcn_wmma_f32_16x16x4_f32(false, a, false, b0,
                                                     (short)0, acc0, false, false);
        acc1 = __builtin_amdgcn_wmma_f32_16x16x4_f32(false, a, false, b1,
                                                     (short)0, acc1, false, false);
        acc2 = __builtin_amdgcn_wmma_f32_16x16x4_f32(false, a, false, b2,
                                                     (short)0, acc2, false, false);
        acc3 = __builtin_amdgcn_wmma_f32_16x16x4_f32(false, a, false, b3,
                                                     (short)0, acc3, false, false);
    }

    // Store D: VGPR v holds rows (m0+v) [lanes 0-15] and (m0+v+8) [lanes 16-31].
    const int rbase = m0 + (khalf << 3);
    #pragma unroll
    for (int v = 0; v < 8; ++v) {
        float* __restrict__ srow = S + (long long)(rbase + v) * DIM + ncol;
        srow[ 0] = acc0[v];
        srow[16] = acc1[v];
        srow[32] = acc2[v];
        srow[48] = acc3[v];
    }
}

// ---------------------------------------------------------------------------
// out[row[e]] += val[e] * support[col[e]]  (COO scatter, L2-resident atomics)
// 64 consecutive lanes = the 64 dims of one edge -> coalesced L2 transactions.
// ---------------------------------------------------------------------------
__global__ void gcn_scatter_kernel(const float* __restrict__ S,
                                   const int* __restrict__ adj_row,
                                   const int* __restrict__ adj_col,
                                   const float* __restrict__ adj_val,
                                   float* __restrict__ out,
                                   int n_edges) {
    const long long idx = (long long)blockIdx.x * blockDim.x + threadIdx.x;
    const int e = (int)(idx >> 6);
    const int d = (int)(idx & 63);
    if (e >= n_edges) return;

    const int   r = adj_row[e];
    const int   c = adj_col[e];
    const float v = adj_val[e];

    atomicAdd(out + (long long)r * DIM + d, v * S[(long long)c * DIM + d]);
}

// ---------------------------------------------------------------------------
extern "C" void kernel_launch(void* const* d_in, const int* in_sizes, int n_in,
                              void* d_out, int out_size, void* d_ws, size_t ws_size,
                              hipStream_t stream) {
    const float* X    = (const float*)d_in[0];   // [N, 64]
    const float* W    = (const float*)d_in[1];   // [64, 64]
    const int*   arow = (const int*)d_in[2];     // [E]
    const int*   acol = (const int*)d_in[3];     // [E]
    const float* aval = (const float*)d_in[4];   // [E]
    float*       out  = (float*)d_out;           // [N, 64]
    float*       S    = (float*)d_ws;            // support scratch [N, 64]

    const int n_nodes = in_sizes[0] / DIM;       // 100000
    const int n_edges = in_sizes[2];             // 1600000

    // 1) zero output
    {
        const int n4 = out_size / 4;
        gcn_zero_kernel<<<(n4 + 255) / 256, 256, 0, stream>>>(out, n4);
    }

    // 2) support = X @ W  (WMMA fp32). N is a multiple of 16 here (100000/16 = 6250).
    {
        const int tiles = n_nodes / 16;
        const int waves_per_block = 256 / 32;    // 8 waves
        const int blocks = (tiles + waves_per_block - 1) / waves_per_block;
        gcn_gemm_wmma_kernel<<<blocks, 256, 0, stream>>>(X, W, S, tiles);
    }

    // 3) COO scatter with f32 atomics (L2-resident: support = 25.6 MB << 192 MB L2)
    {
        const long long work = (long long)n_edges * DIM;
        const int blocks = (int)((work + 255) / 256);
        gcn_scatter_kernel<<<blocks, 256, 0, stream>>>(S, arow, acol, aval, out, n_edges);
    }
}